// HeteroRGCN_28209345200162
// MI455X (gfx1250) — compile-verified
//
#include <hip/hip_runtime.h>

#define D 64

typedef __attribute__((ext_vector_type(2))) float v2f;
typedef __attribute__((ext_vector_type(8))) float v8f;

// Hardware fp32 atomic add (device scope, relaxed) -> global_atomic_add_f32,
// avoiding any CAS-loop fallback in the bandwidth-critical scatter.
__device__ __forceinline__ void fadd_atomic(float* p, float v) {
  __hip_atomic_fetch_add(p, v, __ATOMIC_RELAXED, __HIP_MEMORY_SCOPE_AGENT);
}

// ---------------------------------------------------------------------------
// Degree counting: cnt[dst] += 1.0 per edge (fp32 atomic, L2-resident array)
// ---------------------------------------------------------------------------
__global__ void edge_count_kernel(const int* __restrict__ dst,
                                  float* __restrict__ cnt, int ne) {
  int e = blockIdx.x * blockDim.x + threadIdx.x;
  if (e < ne) fadd_atomic(&cnt[dst[e]], 1.0f);
}

// c[i] = 1 / max(c[i], 1)  (segment-mean denominator; empty segments -> /1)
__global__ void inv_kernel(float* __restrict__ c, int n) {
  int i = blockIdx.x * blockDim.x + threadIdx.x;
  if (i < n) c[i] = 1.0f / fmaxf(c[i], 1.0f);
}

// ---------------------------------------------------------------------------
// Projection GEMM: out[nrows x 64] = in[nrows x 64] @ W[64 x 64] + bias[64]
// One wave32 computes a 16-row x 64-col strip via V_WMMA_F32_16X16X4_F32.
// A (16x4 f32) fragment layout (ISA 7.12.2): lanes 0-15 hold M=lane,K={kk,kk+1};
// lanes 16-31 hold M=lane-16,K={kk+2,kk+3}. B (4x16) mirrors with N across
// lanes. C/D (16x16 f32): VGPR r -> M = r + 8*(lane>=16), N = lane&15.
// ---------------------------------------------------------------------------
__global__ __launch_bounds__(256) void proj_gemm_kernel(
    const float* __restrict__ in, const float* __restrict__ W,
    const float* __restrict__ bias, float* __restrict__ out, int nrows) {
  const int lane = threadIdx.x & 31;
  const int wave = blockIdx.x * (blockDim.x >> 5) + (threadIdx.x >> 5);
  const int row0 = wave << 4;
  if (row0 >= nrows) return;          // wave-uniform: EXEC stays all-ones for WMMA
  const int half = lane >> 4;         // 0: lanes 0-15, 1: lanes 16-31
  const int l16  = lane & 15;

  int ar = row0 + l16;
  if (ar >= nrows) ar = nrows - 1;    // clamp reads; stores guarded below
  const float* arow = in + (size_t)ar * D;

  v8f c0 = {}, c1 = {}, c2 = {}, c3 = {};
#pragma unroll
  for (int kk = 0; kk < D; kk += 4) {
    v2f a;
    a[0] = arow[kk + half * 2 + 0];
    a[1] = arow[kk + half * 2 + 1];
    const float* w0 = W + (size_t)(kk + half * 2) * D;  // row K = kk + 2*half
    const float* w1 = w0 + D;                           // row K = kk + 2*half + 1
    v2f b0v, b1v, b2v, b3v;
    b0v[0] = w0[l16 +  0]; b0v[1] = w1[l16 +  0];
    b1v[0] = w0[l16 + 16]; b1v[1] = w1[l16 + 16];
    b2v[0] = w0[l16 + 32]; b2v[1] = w1[l16 + 32];
    b3v[0] = w0[l16 + 48]; b3v[1] = w1[l16 + 48];
    c0 = __builtin_amdgcn_wmma_f32_16x16x4_f32(false, a, false, b0v, (short)0, c0, false, false);
    c1 = __builtin_amdgcn_wmma_f32_16x16x4_f32(false, a, false, b1v, (short)0, c1, false, false);
    c2 = __builtin_amdgcn_wmma_f32_16x16x4_f32(false, a, false, b2v, (short)0, c2, false, false);
    c3 = __builtin_amdgcn_wmma_f32_16x16x4_f32(false, a, false, b3v, (short)0, c3, false, false);
  }

  const float bi0 = bias[l16], bi1 = bias[l16 + 16],
              bi2 = bias[l16 + 32], bi3 = bias[l16 + 48];
#pragma unroll
  for (int r = 0; r < 8; ++r) {
    int row = row0 + r + half * 8;
    if (row < nrows) {
      float* o = out + (size_t)row * D + l16;
      o[0]  = c0[r] + bi0;
      o[16] = c1[r] + bi1;
      o[32] = c2[r] + bi2;
      o[48] = c3[r] + bi3;
    }
  }
}

// ---------------------------------------------------------------------------
// Edge mean-scatter: acc[dst] += wh[src] * inv_deg[dst]
// 16 threads per edge, float4 (16B) gather per thread, 4 fp32 HW atomics.
// Accumulator arrays are <= 76.8 MB -> L2 (192 MB) resident atomics.
// ---------------------------------------------------------------------------
__global__ void edge_scatter_kernel(const float* __restrict__ wh,
                                    const int* __restrict__ src,
                                    const int* __restrict__ dst,
                                    const float* __restrict__ inv,
                                    float* __restrict__ acc, int ne) {
  int tid = blockIdx.x * blockDim.x + threadIdx.x;
  if (tid >= ne * 16) return;
  int e    = tid >> 4;
  int part = tid & 15;
  int s = src[e], d = dst[e];
  float w = inv[d];
  float4 v = *(const float4*)(wh + (size_t)s * D + part * 4);
  float* a = acc + (size_t)d * D + part * 4;
  fadd_atomic(a + 0, v.x * w);
  fadd_atomic(a + 1, v.y * w);
  fadd_atomic(a + 2, v.z * w);
  fadd_atomic(a + 3, v.w * w);
}

// leaky_relu(x, 0.01) in place
__global__ void leaky_kernel(float* __restrict__ x, int n) {
  int i = blockIdx.x * blockDim.x + threadIdx.x;
  if (i < n) {
    float v = x[i];
    x[i] = v > 0.0f ? v : 0.01f * v;
  }
}

// out[n x 2] = h[n x 64] @ Wc[64 x 2] + bc[2]
__global__ void classifier_kernel(const float* __restrict__ h,
                                  const float* __restrict__ Wc,
                                  const float* __restrict__ bc,
                                  float* __restrict__ out, int n) {
  int i = blockIdx.x * blockDim.x + threadIdx.x;
  if (i >= n) return;
  const float* row = h + (size_t)i * D;
  float a0 = bc[0], a1 = bc[1];
#pragma unroll
  for (int k = 0; k < D; ++k) {
    float v = row[k];
    a0 += v * Wc[k * 2 + 0];
    a1 += v * Wc[k * 2 + 1];
  }
  out[(size_t)i * 2 + 0] = a0;
  out[(size_t)i * 2 + 1] = a1;
}

static inline int cdiv(long long a, long long b) { return (int)((a + b - 1) / b); }

extern "C" void kernel_launch(void* const* d_in, const int* in_sizes, int n_in,
                              void* d_out, int out_size, void* d_ws, size_t ws_size,
                              hipStream_t stream) {
  // setup_inputs() order
  const float* features = (const float*)d_in[0];
  // d_in[1] emb_user, d_in[2] emb_device: dead in the pruned dataflow
  const float* W0 = (const float*)d_in[3];   // (4,64,64), etype order u2t,t2u,d2t,t2d
  const float* b0 = (const float*)d_in[4];   // (4,64)
  const float* W1 = (const float*)d_in[5];
  const float* b1 = (const float*)d_in[6];
  const float* Wc = (const float*)d_in[7];   // (64,2)
  const float* bc = (const float*)d_in[8];   // (2,)
  const int* src_u2t = (const int*)d_in[9];
  const int* dst_u2t = (const int*)d_in[10];
  const int* src_t2u = (const int*)d_in[11];
  const int* dst_t2u = (const int*)d_in[12];
  const int* src_d2t = (const int*)d_in[13];
  const int* dst_d2t = (const int*)d_in[14];
  const int* src_t2d = (const int*)d_in[15];
  const int* dst_t2d = (const int*)d_in[16];
  float* out = (float*)d_out;

  const int NT = in_sizes[0] / D;
  const int NU = in_sizes[1] / D;
  const int ND = in_sizes[2] / D;
  const int E  = in_sizes[9];
  const int WH = NT > NU ? NT : NU;

  // Workspace layout (floats)
  float* inv_t2u = (float*)d_ws;                    // NU
  float* inv_t2d = inv_t2u + NU;                    // ND
  float* inv_u2t = inv_t2d + ND;                    // NT
  float* inv_d2t = inv_u2t + NT;                    // NT
  float* wh      = inv_d2t + NT;                    // WH*D (projection scratch)
  float* hU      = wh + (size_t)WH * D;             // NU*D (layer-0 user out)
  float* hD      = hU + (size_t)NU * D;             // ND*D (layer-0 device out)
  float* hT      = hD + (size_t)ND * D;             // NT*D (layer-1 target out)
  size_t need = (size_t)(NU + ND + 2 * NT) * 4 +
                ((size_t)WH + NU + ND + NT) * D * 4;
  if (ws_size < need) return;  // workspace too small: refuse rather than corrupt

  const int B = 256;
  const int nInv = NU + ND + NT + NT;

  // ---- inverse degrees (shared by both layers) ----
  hipMemsetAsync(inv_t2u, 0, sizeof(float) * (size_t)nInv, stream);
  edge_count_kernel<<<cdiv(E, B), B, 0, stream>>>(dst_t2u, inv_t2u, E);
  edge_count_kernel<<<cdiv(E, B), B, 0, stream>>>(dst_t2d, inv_t2d, E);
  edge_count_kernel<<<cdiv(E, B), B, 0, stream>>>(dst_u2t, inv_u2t, E);
  edge_count_kernel<<<cdiv(E, B), B, 0, stream>>>(dst_d2t, inv_d2t, E);
  inv_kernel<<<cdiv(nInv, B), B, 0, stream>>>(inv_t2u, nInv);

  const int gemmBlocksT = cdiv(cdiv(NT, 16), B / 32);
  const int gemmBlocksU = cdiv(cdiv(NU, 16), B / 32);
  const int gemmBlocksD = cdiv(cdiv(ND, 16), B / 32);
  const int scatGrid    = cdiv((long long)E * 16, B);

  // ---- layer 0 (only live relations: t2u -> hU, t2d -> hD) ----
  proj_gemm_kernel<<<gemmBlocksT, B, 0, stream>>>(features, W0 + 1 * D * D, b0 + 1 * D, wh, NT);
  hipMemsetAsync(hU, 0, sizeof(float) * (size_t)NU * D, stream);
  edge_scatter_kernel<<<scatGrid, B, 0, stream>>>(wh, src_t2u, dst_t2u, inv_t2u, hU, E);

  proj_gemm_kernel<<<gemmBlocksT, B, 0, stream>>>(features, W0 + 3 * D * D, b0 + 3 * D, wh, NT);
  hipMemsetAsync(hD, 0, sizeof(float) * (size_t)ND * D, stream);
  edge_scatter_kernel<<<scatGrid, B, 0, stream>>>(wh, src_t2d, dst_t2d, inv_t2d, hD, E);

  // leaky_relu on hU,hD (contiguous region)
  leaky_kernel<<<cdiv((long long)(NU + ND) * D, B), B, 0, stream>>>(hU, (NU + ND) * D);

  // ---- layer 1 (only new_t needed: u2t + d2t -> hT) ----
  proj_gemm_kernel<<<gemmBlocksU, B, 0, stream>>>(hU, W1 + 0 * D * D, b1 + 0 * D, wh, NU);
  hipMemsetAsync(hT, 0, sizeof(float) * (size_t)NT * D, stream);
  edge_scatter_kernel<<<scatGrid, B, 0, stream>>>(wh, src_u2t, dst_u2t, inv_u2t, hT, E);

  proj_gemm_kernel<<<gemmBlocksD, B, 0, stream>>>(hD, W1 + 2 * D * D, b1 + 2 * D, wh, ND);
  edge_scatter_kernel<<<scatGrid, B, 0, stream>>>(wh, src_d2t, dst_d2t, inv_d2t, hT, E);

  // ---- classifier ----
  classifier_kernel<<<cdiv(NT, B), B, 0, stream>>>(hT, Wc, bc, out, NT);
}